// QuantizedLinear_84550726189331
// MI455X (gfx1250) — compile-verified
//
#include <hip/hip_runtime.h>

typedef __attribute__((ext_vector_type(2))) float v2f;
typedef __attribute__((ext_vector_type(8))) float v8f;

#define BM 128
#define BN 128
#define BK 128                 // == GROUP_SIZE: one quant group per K-tile
#define LDS_PITCH (BK + 4)     // pad to dodge bank conflicts; keeps 16B alignment
#define GROUP_SIZE 128

// ---------------------------------------------------------------------------
// Pass 1: group-wise int4 fake-quant of W, one wave32 per 128-element group.
// Fully coalesced float4 traffic, 5-step butterfly absmax reduction.
// ---------------------------------------------------------------------------
__global__ __launch_bounds__(256)
void dequant_w_kernel(const float* __restrict__ w, float* __restrict__ wq)
{
    const int g    = blockIdx.x * 8 + (threadIdx.x >> 5);  // group index
    const int lane = threadIdx.x & 31;

    const float4* src = (const float4*)(w + (size_t)g * GROUP_SIZE) + lane;
    float4 v = *src;

    float amax = fmaxf(fmaxf(fabsf(v.x), fabsf(v.y)), fmaxf(fabsf(v.z), fabsf(v.w)));
#pragma unroll
    for (int off = 16; off >= 1; off >>= 1)
        amax = fmaxf(amax, __shfl_xor(amax, off, 32));

    const float scale = amax * (1.0f / 7.0f);
    const float inv   = (scale > 0.0f) ? (1.0f / scale) : 0.0f;  // scale==0 -> q=0 -> wq=0

    float4 q;
    q.x = fminf(fmaxf(rintf(v.x * inv), -8.0f), 7.0f) * scale;
    q.y = fminf(fmaxf(rintf(v.y * inv), -8.0f), 7.0f) * scale;
    q.z = fminf(fmaxf(rintf(v.z * inv), -8.0f), 7.0f) * scale;
    q.w = fminf(fmaxf(rintf(v.w * inv), -8.0f), 7.0f) * scale;

    ((float4*)(wq + (size_t)g * GROUP_SIZE))[lane] = q;
}

// ---------------------------------------------------------------------------
// Pass 2: GEMM  y = x @ wq^T + bias  with V_WMMA_F32_16X16X4_F32.
// FUSED=true keeps the in-kernel dequant (fallback when d_ws is too small).
// grid = (N/BN, M/BM), block = 256 threads (8 wave32)
// ---------------------------------------------------------------------------
template <bool FUSED>
__global__ __launch_bounds__(256)
void qlinear_wmma_f32_kernel(const float* __restrict__ x,
                             const float* __restrict__ w,
                             const float* __restrict__ bias,
                             float* __restrict__ out,
                             int M, int N, int K)
{
    extern __shared__ float lds[];
    float* As = lds;                      // [BM][LDS_PITCH]  x tile
    float* Ws = lds + BM * LDS_PITCH;     // [BN][LDS_PITCH]  (dequantized) w tile

    const int tid  = threadIdx.x;
    const int lane = tid & 31;
    const int wave = tid >> 5;            // 0..7
    const int wrow = wave & 3;            // 32-row strip
    const int wcol = wave >> 2;           // 64-col strip

    const int mb = blockIdx.y * BM;
    const int nb = blockIdx.x * BN;

    // cooperative loader: 2 threads per tile-row, 64 contiguous floats each
    const int lrow  = tid >> 1;           // 0..127
    const int lhalf = (tid & 1) * 64;     // 0 or 64

    const int l16 = lane & 15;
    const int khi = (lane >> 4) << 1;     // 0 or 2 : K-pair base within x4 step

    v8f acc[2][4];
#pragma unroll
    for (int i = 0; i < 2; ++i)
#pragma unroll
        for (int j = 0; j < 4; ++j)
            acc[i][j] = (v8f)(0.0f);

    for (int kt = 0; kt < K; kt += BK) {
        // ---------------- load x tile -> LDS ----------------
        {
            const float4* src = (const float4*)(x + (size_t)(mb + lrow) * K + kt + lhalf);
            float4* dst = (float4*)(As + lrow * LDS_PITCH + lhalf);
#pragma unroll
            for (int i = 0; i < 16; ++i) dst[i] = src[i];
            if (kt + BK < K)
                __builtin_prefetch(x + (size_t)(mb + lrow) * K + kt + BK + lhalf, 0, 0);
        }
        // ---------------- load w tile -> LDS ----------------
        if constexpr (FUSED) {
            // in-kernel group fake-quant (used only when no workspace available)
            const float4* src = (const float4*)(w + (size_t)(nb + lrow) * K + kt + lhalf);
            float4 v[16];
            float amax = 0.0f;
#pragma unroll
            for (int i = 0; i < 16; ++i) {
                v[i] = src[i];
                amax = fmaxf(amax, fmaxf(fmaxf(fabsf(v[i].x), fabsf(v[i].y)),
                                         fmaxf(fabsf(v[i].z), fabsf(v[i].w))));
            }
            amax = fmaxf(amax, __shfl_xor(amax, 1, 32));
            const float scale = amax * (1.0f / 7.0f);
            const float inv   = (scale > 0.0f) ? (1.0f / scale) : 0.0f;
            float4* dst = (float4*)(Ws + lrow * LDS_PITCH + lhalf);
#pragma unroll
            for (int i = 0; i < 16; ++i) {
                float4 q;
                q.x = fminf(fmaxf(rintf(v[i].x * inv), -8.0f), 7.0f) * scale;
                q.y = fminf(fmaxf(rintf(v[i].y * inv), -8.0f), 7.0f) * scale;
                q.z = fminf(fmaxf(rintf(v[i].z * inv), -8.0f), 7.0f) * scale;
                q.w = fminf(fmaxf(rintf(v[i].w * inv), -8.0f), 7.0f) * scale;
                dst[i] = q;
            }
        } else {
            // w already dequantized by pass 1: straight copy
            const float4* src = (const float4*)(w + (size_t)(nb + lrow) * K + kt + lhalf);
            float4* dst = (float4*)(Ws + lrow * LDS_PITCH + lhalf);
#pragma unroll
            for (int i = 0; i < 16; ++i) dst[i] = src[i];
        }
        if (kt + BK < K)
            __builtin_prefetch(w + (size_t)(nb + lrow) * K + kt + BK + lhalf, 0, 0);
        __syncthreads();

        // ---------------- WMMA f32 16x16x4 over this K-tile ----------------
        // A frag layout: lane%16 = M row, VGPR0/1 hold K = khi, khi+1
        const float* Abase = As + (wrow * 32 + l16) * LDS_PITCH + khi;
        // B frag layout (transposed analog): lane%16 = N col of w-tile row
        const float* Bbase = Ws + (wcol * 64 + l16) * LDS_PITCH + khi;

#pragma unroll 4
        for (int kk = 0; kk < BK; kk += 4) {
            v2f a0 = *(const v2f*)(Abase + kk);
            v2f a1 = *(const v2f*)(Abase + 16 * LDS_PITCH + kk);
            v2f b0 = *(const v2f*)(Bbase + kk);
            v2f b1 = *(const v2f*)(Bbase + 16 * LDS_PITCH + kk);
            v2f b2 = *(const v2f*)(Bbase + 32 * LDS_PITCH + kk);
            v2f b3 = *(const v2f*)(Bbase + 48 * LDS_PITCH + kk);

            acc[0][0] = __builtin_amdgcn_wmma_f32_16x16x4_f32(false, a0, false, b0, (short)0, acc[0][0], false, false);
            acc[0][1] = __builtin_amdgcn_wmma_f32_16x16x4_f32(false, a0, false, b1, (short)0, acc[0][1], false, false);
            acc[0][2] = __builtin_amdgcn_wmma_f32_16x16x4_f32(false, a0, false, b2, (short)0, acc[0][2], false, false);
            acc[0][3] = __builtin_amdgcn_wmma_f32_16x16x4_f32(false, a0, false, b3, (short)0, acc[0][3], false, false);
            acc[1][0] = __builtin_amdgcn_wmma_f32_16x16x4_f32(false, a1, false, b0, (short)0, acc[1][0], false, false);
            acc[1][1] = __builtin_amdgcn_wmma_f32_16x16x4_f32(false, a1, false, b1, (short)0, acc[1][1], false, false);
            acc[1][2] = __builtin_amdgcn_wmma_f32_16x16x4_f32(false, a1, false, b2, (short)0, acc[1][2], false, false);
            acc[1][3] = __builtin_amdgcn_wmma_f32_16x16x4_f32(false, a1, false, b3, (short)0, acc[1][3], false, false);
        }
        __syncthreads();
    }

    // ---------------- epilogue: bias add + store ----------------
    // C/D layout: VGPR v holds M = v (lanes 0-15) and M = v+8 (lanes 16-31), N = lane%16
#pragma unroll
    for (int ct = 0; ct < 4; ++ct) {
        const int n  = nb + wcol * 64 + ct * 16 + l16;
        const float bv = bias[n];
#pragma unroll
        for (int rt = 0; rt < 2; ++rt) {
            const int m0 = mb + wrow * 32 + rt * 16 + ((lane >> 4) << 3);
#pragma unroll
            for (int v = 0; v < 8; ++v) {
                out[(size_t)(m0 + v) * N + n] = acc[rt][ct][v] + bv;
            }
        }
    }
}

extern "C" void kernel_launch(void* const* d_in, const int* in_sizes, int n_in,
                              void* d_out, int out_size, void* d_ws, size_t ws_size,
                              hipStream_t stream) {
    const float* x    = (const float*)d_in[0];   // [B*S, K] fp32
    const float* w    = (const float*)d_in[1];   // [N, K]   fp32
    const float* bias = (const float*)d_in[2];   // [N]      fp32
    float* out = (float*)d_out;                  // [B*S, N] fp32

    const int N = in_sizes[2];
    const int K = in_sizes[1] / N;
    const int M = in_sizes[0] / K;

    const size_t shmem = (size_t)2 * BM * LDS_PITCH * sizeof(float);  // ~135 KB < 320 KB WGP LDS
    dim3 grid(N / BN, M / BM);

    const size_t wq_bytes = (size_t)in_sizes[1] * sizeof(float);
    if (ws_size >= wq_bytes) {
        // Pass 1: dequantize W once (removes ~6.4G redundant VALU lane-ops)
        float* wq = (float*)d_ws;
        const int groups = in_sizes[1] / GROUP_SIZE;     // one wave per group
        dequant_w_kernel<<<groups / 8, 256, 0, stream>>>(w, wq);

        (void)hipFuncSetAttribute((const void*)qlinear_wmma_f32_kernel<false>,
                                  hipFuncAttributeMaxDynamicSharedMemorySize, (int)shmem);
        qlinear_wmma_f32_kernel<false><<<grid, 256, shmem, stream>>>(x, wq, bias, out, M, N, K);
    } else {
        (void)hipFuncSetAttribute((const void*)qlinear_wmma_f32_kernel<true>,
                                  hipFuncAttributeMaxDynamicSharedMemorySize, (int)shmem);
        qlinear_wmma_f32_kernel<true><<<grid, 256, shmem, stream>>>(x, w, bias, out, M, N, K);
    }
}